// GAT_layer_76785425318241
// MI455X (gfx1250) — compile-verified
//
#include <hip/hip_runtime.h>

// ---------------- problem constants (match reference) ----------------
#define N_NODES   50000
#define N_EDGES   800000
#define IN_DIM    256
#define OUT_DIM   64
#define NEG_SLOPE 0.01f

typedef __attribute__((ext_vector_type(2))) float v2f;
typedef __attribute__((ext_vector_type(8))) float v8f;

// ---------------------------------------------------------------------
// Kernel 1: init workspace (agg=0, denom=0, emax=-inf)
// ---------------------------------------------------------------------
__global__ __launch_bounds__(256) void gat_init(float* __restrict__ agg,
                                                float* __restrict__ denom,
                                                float* __restrict__ emax) {
    int i = blockIdx.x * blockDim.x + threadIdx.x;
    if (i < N_NODES * OUT_DIM) agg[i] = 0.0f;
    if (i < N_NODES) {
        denom[i] = 0.0f;
        emax[i]  = -__builtin_inff();
    }
}

// ---------------------------------------------------------------------
// Kernel 2: h = hidden @ W + b  via V_WMMA_F32_16X16X4_F32 (wave32).
// Block = 128 threads = 4 waves; wave w owns the 16-wide N-tile w (OUT_DIM=64).
// Each block covers 16 rows of `hidden`; grid.x = 50000/16 = 3125.
//
// Operand layouts per CDNA5 ISA 7.12.2 (32-bit A 16x4 / B 4x16 / C 16x16):
//   A: lanes 0-15 hold row M=lane, VGPR0=K0,VGPR1=K1; lanes 16-31 hold K2,K3
//   B: lanes 0-15 hold col N=lane, rows K0/K1; lanes 16-31 rows K2/K3
//   C: VGPR r -> (M=r, N=lane) for lanes 0-15, (M=8+r, N=lane-16) for 16-31
// ---------------------------------------------------------------------
__global__ __launch_bounds__(128) void gat_gemm_wmma(
    const float* __restrict__ hidden,   // [N, 256]
    const float* __restrict__ W,        // [256, 64] row-major
    const float* __restrict__ bias,     // [64]
    float* __restrict__ h_out)          // [N, 64]
{
    const int lane  = threadIdx.x & 31;
    const int wave  = threadIdx.x >> 5;     // 0..3 -> which 16-col tile
    const int half  = lane >> 4;            // 0: K+0/1 side, 1: K+2/3 side
    const int l16   = lane & 15;
    const int mbase = blockIdx.x * 16;
    const int gcol  = wave * 16 + l16;

    // per-lane base pointers
    const float* arow = hidden + (size_t)(mbase + l16) * IN_DIM + 2 * half;
    const float* bcol = W + (size_t)(2 * half) * OUT_DIM + gcol;

    v8f acc = {};
#pragma unroll 4
    for (int k0 = 0; k0 < IN_DIM; k0 += 4) {
        // A fragment: two consecutive K values for this lane's row (8B aligned)
        v2f a = *(const v2f*)(arow + k0);
        // B fragment: two consecutive K rows of column gcol (stride OUT_DIM)
        v2f b;
        b.x = bcol[(size_t)k0 * OUT_DIM];
        b.y = bcol[(size_t)(k0 + 1) * OUT_DIM];
        acc = __builtin_amdgcn_wmma_f32_16x16x4_f32(
            /*neg_a=*/false, a, /*neg_b=*/false, b,
            /*c_mod=*/(short)0, acc, /*reuse_a=*/false, /*reuse_b=*/false);
    }

    const float bb = bias[gcol];
#pragma unroll
    for (int r = 0; r < 8; ++r) {
        int row = mbase + r + 8 * half;
        h_out[(size_t)row * OUT_DIM + gcol] = acc[r] + bb;
    }
}

// ---------------------------------------------------------------------
// Kernel 3: per-node attention projections
//   a_src[i] = h[i,:] . att_w[0:64],  a_dst[i] = h[i,:] . att_w[64:128]
// One wave per node, 2 features per lane, shuffle reduction.
// ---------------------------------------------------------------------
__global__ __launch_bounds__(256) void gat_proj(
    const float* __restrict__ h,        // [N, 64]
    const float* __restrict__ att_w,    // [128]
    float* __restrict__ a_src,
    float* __restrict__ a_dst)
{
    int node = blockIdx.x * 8 + (threadIdx.x >> 5);
    if (node >= N_NODES) return;
    int lane = threadIdx.x & 31;

    float h0 = h[(size_t)node * OUT_DIM + lane];
    float h1 = h[(size_t)node * OUT_DIM + lane + 32];
    float s = h0 * att_w[lane]      + h1 * att_w[lane + 32];
    float d = h0 * att_w[lane + 64] + h1 * att_w[lane + 96];
#pragma unroll
    for (int off = 16; off > 0; off >>= 1) {
        s += __shfl_down(s, off, 32);
        d += __shfl_down(d, off, 32);
    }
    if (lane == 0) { a_src[node] = s; a_dst[node] = d; }
}

// ---------------------------------------------------------------------
// float atomic max via signed-max / unsigned-min trick (valid for any
// sign mix; -inf identity behaves correctly in both orderings).
// ---------------------------------------------------------------------
__device__ __forceinline__ void atomicMaxF(float* addr, float v) {
    if (v >= 0.0f) atomicMax((int*)addr, __float_as_int(v));
    else           atomicMin((unsigned int*)addr, __float_as_uint(v));
}

// ---------------------------------------------------------------------
// Kernel 4: edge logits + segment max over dst
// ---------------------------------------------------------------------
__global__ __launch_bounds__(256) void gat_edge_logit(
    const int* __restrict__ src, const int* __restrict__ dst,
    const float* __restrict__ a_src, const float* __restrict__ a_dst,
    const float* __restrict__ att_b,
    float* __restrict__ e_out, float* __restrict__ emax)
{
    int e = blockIdx.x * blockDim.x + threadIdx.x;
    if (e >= N_EDGES) return;
    int s = src[e], d = dst[e];
    float v = a_src[s] + a_dst[d] + att_b[0];
    v = (v > 0.0f) ? v : NEG_SLOPE * v;      // leaky_relu
    e_out[e] = v;
    atomicMaxF(&emax[d], v);
}

// ---------------------------------------------------------------------
// Kernel 5: softmax numerator + weighted scatter of h[src] into agg[dst].
// One wave per edge; lane c handles features c and c+32 -> coalesced
// 128B gather + two coalesced atomic bursts per edge.
// ---------------------------------------------------------------------
__global__ __launch_bounds__(256) void gat_edge_scatter(
    const int* __restrict__ src, const int* __restrict__ dst,
    const float* __restrict__ e_in, const float* __restrict__ emax,
    const float* __restrict__ h,
    float* __restrict__ denom, float* __restrict__ agg)
{
    int edge = blockIdx.x * 8 + (threadIdx.x >> 5);
    if (edge >= N_EDGES) return;
    int lane = threadIdx.x & 31;
    int s = src[edge], d = dst[edge];

    float ex = __expf(e_in[edge] - emax[d]);   // e - emax <= 0, safe range
    if (lane == 0) atomicAdd(&denom[d], ex);

    float h0 = h[(size_t)s * OUT_DIM + lane];
    float h1 = h[(size_t)s * OUT_DIM + lane + 32];
    atomicAdd(&agg[(size_t)d * OUT_DIM + lane],       ex * h0);
    atomicAdd(&agg[(size_t)d * OUT_DIM + lane + 32],  ex * h1);
}

// ---------------------------------------------------------------------
// Kernel 6: out = denom>0 ? agg/denom : h   (zero-in-degree fallback)
// ---------------------------------------------------------------------
__global__ __launch_bounds__(256) void gat_final(
    const float* __restrict__ agg, const float* __restrict__ denom,
    const float* __restrict__ h, float* __restrict__ out)
{
    int i = blockIdx.x * blockDim.x + threadIdx.x;
    if (i >= N_NODES * OUT_DIM) return;
    int node = i >> 6;
    float dn = denom[node];
    out[i] = (dn > 0.0f) ? agg[i] / dn : h[i];
}

// ---------------------------------------------------------------------
extern "C" void kernel_launch(void* const* d_in, const int* in_sizes, int n_in,
                              void* d_out, int out_size, void* d_ws, size_t ws_size,
                              hipStream_t stream) {
    const float* hidden = (const float*)d_in[0];   // [50000,256]
    const float* W      = (const float*)d_in[1];   // [256,64]
    const float* bias   = (const float*)d_in[2];   // [64]
    const float* att_w  = (const float*)d_in[3];   // [128,1]
    const float* att_b  = (const float*)d_in[4];   // [1]
    const int*   src    = (const int*)  d_in[5];   // [800000]
    const int*   dst    = (const int*)  d_in[6];   // [800000]
    float* out = (float*)d_out;                    // [50000,64]

    // workspace carve-out (~29.6 MB of floats)
    float* ws     = (float*)d_ws;
    float* h_ws   = ws;                               // N*64
    float* a_src  = h_ws  + (size_t)N_NODES * OUT_DIM;
    float* a_dst  = a_src + N_NODES;
    float* emax   = a_dst + N_NODES;
    float* denom  = emax  + N_NODES;
    float* e_ws   = denom + N_NODES;                  // E
    float* agg    = e_ws  + N_EDGES;                  // N*64

    const int NT = 256;
    // 1) init agg/denom/emax
    gat_init<<<(N_NODES * OUT_DIM + NT - 1) / NT, NT, 0, stream>>>(agg, denom, emax);
    // 2) dense GEMM on WMMA (f32 16x16x4): 3125 row-tiles x 4 col-waves
    gat_gemm_wmma<<<N_NODES / 16, 128, 0, stream>>>(hidden, W, bias, h_ws);
    // 3) per-node attention projections
    gat_proj<<<(N_NODES + 7) / 8, NT, 0, stream>>>(h_ws, att_w, a_src, a_dst);
    // 4) edge logits + segment max
    gat_edge_logit<<<(N_EDGES + NT - 1) / NT, NT, 0, stream>>>(
        src, dst, a_src, a_dst, att_b, e_ws, emax);
    // 5) softmax numerators + weighted scatter (wave per edge)
    gat_edge_scatter<<<(N_EDGES + 7) / 8, NT, 0, stream>>>(
        src, dst, e_ws, emax, h_ws, denom, agg);
    // 6) normalize / fallback
    gat_final<<<(N_NODES * OUT_DIM + NT - 1) / NT, NT, 0, stream>>>(
        agg, denom, h_ws, out);
}